// WaveletUnit_50697793962808
// MI455X (gfx1250) — compile-verified
//
#include <hip/hip_runtime.h>
#include <math.h>

#define BATCH 8
#define C64   64
#define INCH  192
#define HID   384
#define H2    128
#define W2    128
#define PB    (H2*W2)          // 16384 positions per batch image
#define PTOT  (BATCH*PB)       // 131072 total spatial positions
#define NB0   (PTOT/64)        // 2048 transpose blocks
#define NTILE (PTOT/16)        // 8192 N tiles (power of two: 2^13)

typedef __attribute__((ext_vector_type(2))) float v2f;
typedef __attribute__((ext_vector_type(8))) float v8f;

// -------------------------------------------------------------------------
// K0: DWT + NHWC->planar transpose + deterministic per-block BN partials
// block = 256 threads handles 64 positions x 64 channels (all 4 bands)
// -------------------------------------------------------------------------
__global__ void __launch_bounds__(256) k_dwt_stats(const float* __restrict__ x,
        float* __restrict__ LL, float* __restrict__ high,
        float* __restrict__ psum, float* __restrict__ psq) {
    __shared__ float tile[256 * 65];        // [plane q][pos pl], pitch 65 (no bank conflicts)
    __shared__ float redS[4 * INCH];
    __shared__ float redQ[4 * INCH];
    const int t  = threadIdx.x;
    const int p0 = blockIdx.x * 64;
    const int c  = t & 63;                  // fixed channel per thread
    float sLH = 0.f, qLH = 0.f, sHL = 0.f, qHL = 0.f, sHH = 0.f, qHH = 0.f;
    for (int it = 0; it < 16; ++it) {
        int pl  = it * 4 + (t >> 6);
        int p   = p0 + pl;
        int b   = p >> 14;
        int rem = p & 16383;
        int y   = rem >> 7;
        int xq  = rem & 127;
        const float* base = x + ((((size_t)b * 256 + 2 * y) * 256 + 2 * xq) * 64 + c);
        float a  = base[0];
        float bb = base[64];                // (2y, 2x+1)
        float cc = base[16384];             // (2y+1, 2x)
        float dd = base[16384 + 64];        // (2y+1, 2x+1)
        float vLL = (a + bb + cc + dd) * 0.5f;
        float vLH = (a - bb + cc - dd) * 0.5f;
        float vHL = (a + bb - cc - dd) * 0.5f;
        float vHH = (a - bb - cc + dd) * 0.5f;
        tile[(c      ) * 65 + pl] = vLL;
        tile[(64  + c) * 65 + pl] = vLH;
        tile[(128 + c) * 65 + pl] = vHL;
        tile[(192 + c) * 65 + pl] = vHH;
        sLH += vLH; qLH += vLH * vLH;
        sHL += vHL; qHL += vHL * vHL;
        sHH += vHH; qHH += vHH * vHH;
    }
    const int g = t >> 6;
    redS[g * INCH +       c] = sLH;  redQ[g * INCH +       c] = qLH;
    redS[g * INCH + 64  + c] = sHL;  redQ[g * INCH + 64  + c] = qHL;
    redS[g * INCH + 128 + c] = sHH;  redQ[g * INCH + 128 + c] = qHH;
    __syncthreads();
    if (t < INCH) {   // deterministic fixed-order partial per (channel, block)
        float s = redS[t] + redS[INCH + t] + redS[2 * INCH + t] + redS[3 * INCH + t];
        float q = redQ[t] + redQ[INCH + t] + redQ[2 * INCH + t] + redQ[3 * INCH + t];
        psum[(size_t)t * NB0 + blockIdx.x] = s;
        psq [(size_t)t * NB0 + blockIdx.x] = q;
    }
    // phase 2: coalesced planar writes (consecutive threads -> consecutive p)
    for (int it = 0; it < 64; ++it) {
        int q  = it * 4 + (t >> 6);
        int pl = t & 63;
        float v = tile[q * 65 + pl];
        if (q < 64) LL  [(size_t)q        * PTOT + p0 + pl] = v;
        else        high[(size_t)(q - 64) * PTOT + p0 + pl] = v;
    }
}

// -------------------------------------------------------------------------
// K1: finalize BN: one block per channel, deterministic tree reduction
// -------------------------------------------------------------------------
__global__ void __launch_bounds__(256) k_bn_finalize(const float* __restrict__ psum,
        const float* __restrict__ psq, const float* __restrict__ gamma,
        const float* __restrict__ beta, float* __restrict__ scale,
        float* __restrict__ bias) {
    __shared__ float s1[256], s2[256];
    const int ch = blockIdx.x;
    const int t  = threadIdx.x;
    float a = 0.f, b = 0.f;
    for (int i = t; i < NB0; i += 256) {
        a += psum[(size_t)ch * NB0 + i];
        b += psq [(size_t)ch * NB0 + i];
    }
    s1[t] = a; s2[t] = b;
    __syncthreads();
    for (int s = 128; s > 0; s >>= 1) {
        if (t < s) { s1[t] += s1[t + s]; s2[t] += s2[t + s]; }
        __syncthreads();
    }
    if (t == 0) {
        float inv  = 1.0f / (float)PTOT;
        float mean = s1[0] * inv;
        float var  = s2[0] * inv - mean * mean;
        float rstd = rsqrtf(var + 1e-5f);
        float sc   = rstd * gamma[ch];
        scale[ch]  = sc;
        bias[ch]   = beta[ch] - mean * sc;
    }
}

// -------------------------------------------------------------------------
// K2/K4: fp32 WMMA GEMM  D[M][P] (=) / (+= res_scale*)  W[M][K] x B[K][P]
// wave = 4 M-tiles x 1 N-tile, K-loop of v_wmma_f32_16x16x4_f32
// -------------------------------------------------------------------------
template<bool NORM_IN, bool ADD_RES>
__global__ void __launch_bounds__(256) k_gemm_wmma(const float* __restrict__ W,
        const float* __restrict__ Bsrc, float* __restrict__ D,
        const float* __restrict__ scale, const float* __restrict__ bias,
        const float* __restrict__ res_scale, int M, int K) {
    const int t    = threadIdx.x;
    const int lane = t & 31;
    const int wid  = t >> 5;
    const long gid = (long)blockIdx.x * 8 + wid;
    const int ntile = (int)(gid & (NTILE - 1));
    const int mg    = (int)(gid >> 13);
    const int n0    = ntile * 16;
    const int m0    = mg * 64;
    const int nl    = lane & 15;            // B col / D col, A row
    const int klo   = (lane >> 4) * 2;      // lanes 0-15 hold K=0,1 ; 16-31 hold K=2,3
    const int rowoff = (lane >> 4) * 8;     // D row offset

    v8f acc[4] = {};
    for (int k0 = 0; k0 < K; k0 += 4) {
        const int ka = k0 + klo;
        float bx = Bsrc[(size_t)ka       * PTOT + n0 + nl];
        float by = Bsrc[(size_t)(ka + 1) * PTOT + n0 + nl];
        if (NORM_IN) {
            bx = fmaf(bx, scale[ka],     bias[ka]);
            by = fmaf(by, scale[ka + 1], bias[ka + 1]);
        }
        v2f bv; bv.x = bx; bv.y = by;
        const float* wp = W + (size_t)(m0 + nl) * K + ka;
#pragma unroll
        for (int mt = 0; mt < 4; ++mt) {
            v2f av;
            av.x = wp[(size_t)(mt * 16) * K];
            av.y = wp[(size_t)(mt * 16) * K + 1];
#if defined(__AMDGCN__)
            acc[mt] = __builtin_amdgcn_wmma_f32_16x16x4_f32(
                false, av, false, bv, (short)0, acc[mt], false, false);
#else
            acc[mt][0] += av.x * bv.x;   // host-pass placeholder, never executed
#endif
        }
    }
    const float rs = ADD_RES ? res_scale[0] : 0.0f;
#pragma unroll
    for (int mt = 0; mt < 4; ++mt) {
#pragma unroll
        for (int r = 0; r < 8; ++r) {
            size_t idx = (size_t)(m0 + mt * 16 + rowoff + r) * PTOT + n0 + nl;
            float v = acc[mt][r];
            if (ADD_RES) D[idx] = fmaf(rs, v, D[idx]);
            else         D[idx] = v;
        }
    }
}

// -------------------------------------------------------------------------
// K3: depthwise 3x3 (pad 1) + exact GELU, planar [HID][P] -> [HID][P]
// -------------------------------------------------------------------------
__global__ void __launch_bounds__(256) k_dw_gelu(const float* __restrict__ hin,
        const float* __restrict__ wdw, float* __restrict__ hout) {
    const int o   = blockIdx.x >> 9;                         // PTOT/256 = 512 blocks/plane
    const int p   = ((blockIdx.x & 511) << 8) + threadIdx.x;
    const int b   = p >> 14;
    const int rem = p & 16383;
    const int y   = rem >> 7;
    const int xq  = rem & 127;
    const float* wp    = wdw + o * 9;
    const float* plane = hin + (size_t)o * PTOT + (size_t)b * PB;
    float s = 0.f;
#pragma unroll
    for (int ky = 0; ky < 3; ++ky) {
        int yy = y + ky - 1;
        if (yy < 0 || yy >= H2) continue;
#pragma unroll
        for (int kx = 0; kx < 3; ++kx) {
            int xx = xq + kx - 1;
            if (xx < 0 || xx >= W2) continue;
            s = fmaf(wp[ky * 3 + kx], plane[yy * W2 + xx], s);
        }
    }
    float g = 0.5f * s * (1.0f + erff(s * 0.70710678118654752f));
    hout[(size_t)o * PTOT + p] = g;
}

// -------------------------------------------------------------------------
// K5: pixel-shuffle "_iwt": LL->(2y,2x), LH->(2y,2x+1), HL->(2y+1,2x),
//     HH->(2y+1,2x+1); planar -> NHWC via LDS transpose
// -------------------------------------------------------------------------
__global__ void __launch_bounds__(256) k_iwt_out(const float* __restrict__ LL,
        const float* __restrict__ high, float* __restrict__ out) {
    __shared__ float tile[256 * 65];
    const int t  = threadIdx.x;
    const int p0 = blockIdx.x * 64;
    for (int it = 0; it < 64; ++it) {
        int q  = it * 4 + (t >> 6);
        int pl = t & 63;
        float v = (q < 64) ? LL  [(size_t)q        * PTOT + p0 + pl]
                           : high[(size_t)(q - 64) * PTOT + p0 + pl];
        tile[q * 65 + pl] = v;
    }
    __syncthreads();
    const int c = t & 63;
    for (int it = 0; it < 16; ++it) {
        int pl  = it * 4 + (t >> 6);
        int p   = p0 + pl;
        int b   = p >> 14;
        int rem = p & 16383;
        int y   = rem >> 7;
        int xq  = rem & 127;
        float vLL = tile[(c      ) * 65 + pl];
        float vLH = tile[(64  + c) * 65 + pl];
        float vHL = tile[(128 + c) * 65 + pl];
        float vHH = tile[(192 + c) * 65 + pl];
        size_t base = (((size_t)b * 256 + 2 * y) * 256 + 2 * xq) * 64 + c;
        out[base]               = vLL;
        out[base + 64]          = vLH;
        out[base + 16384]       = vHL;
        out[base + 16384 + 64]  = vHH;
    }
}

// -------------------------------------------------------------------------
extern "C" void kernel_launch(void* const* d_in, const int* in_sizes, int n_in,
                              void* d_out, int out_size, void* d_ws, size_t ws_size,
                              hipStream_t stream) {
    const float* x         = (const float*)d_in[0];
    const float* gamma     = (const float*)d_in[1];
    const float* beta      = (const float*)d_in[2];
    const float* w_in      = (const float*)d_in[3];
    const float* w_dw      = (const float*)d_in[4];
    const float* w_out     = (const float*)d_in[5];
    const float* res_scale = (const float*)d_in[6];

    float* psum  = (float*)d_ws;                         // [192][2048]
    float* psq   = psum  + (size_t)INCH * NB0;           // [192][2048]
    float* scale = psq   + (size_t)INCH * NB0;           // [192]
    float* bias  = scale + INCH;                         // [192]
    float* LL    = bias  + INCH;                         // [64][P]   32 MB
    float* high  = LL    + (size_t)C64  * PTOT;          // [192][P]  96 MB
    float* hid1  = high  + (size_t)INCH * PTOT;          // [384][P] 192 MB
    float* hid2  = hid1  + (size_t)HID  * PTOT;          // [384][P] 192 MB
    float* out   = (float*)d_out;

    k_dwt_stats  <<<NB0,  256, 0, stream>>>(x, LL, high, psum, psq);
    k_bn_finalize<<<INCH, 256, 0, stream>>>(psum, psq, gamma, beta, scale, bias);

    // GEMM1: hid1[384][P] = w_in x BN(high);  8192 N-tiles * 6 M-groups / 8 waves
    k_gemm_wmma<true, false><<<NTILE * (HID / 64) / 8, 256, 0, stream>>>(
        w_in, high, hid1, scale, bias, res_scale, HID, INCH);

    k_dw_gelu<<<HID * (PTOT / 256), 256, 0, stream>>>(hid1, w_dw, hid2);

    // GEMM2: high[192][P] += res_scale * (w_out x hid2)   (in-place residual)
    k_gemm_wmma<false, true><<<NTILE * (INCH / 64) / 8, 256, 0, stream>>>(
        w_out, hid2, high, scale, bias, res_scale, INCH, HID);

    k_iwt_out<<<NB0, 256, 0, stream>>>(LL, high, out);
}